// _HANGBlock_18047452578207
// MI455X (gfx1250) — compile-verified
//
#include <hip/hip_runtime.h>

#define HIDDEN 96
#define NN 50000
#define NE 800000

typedef __attribute__((ext_vector_type(2))) float v2f;
typedef __attribute__((ext_vector_type(8))) float v8f;

__device__ __forceinline__ v8f wmma_f32_k4(v2f a, v2f b, v8f c) {
  // V_WMMA_F32_16X16X4_F32 : D(16x16 f32) = A(16x4 f32) * B(4x16 f32) + C
  return __builtin_amdgcn_wmma_f32_16x16x4_f32(false, a, false, b, (short)0, c,
                                               false, false);
}

// ---- dual-tile 16x16 GEMM helpers ----------------------------------------
// A (16 x K) in LDS, row-major, leading dim lda. Two independent accumulator
// chains (c0,c1) share each A-fragment: 2x WMMA ILP, half the A loads.
// A-frag layout (ISA 32-bit A 16x4): lanes 0-15 -> K={k,k+1}, lanes 16-31 ->
// K={k+2,k+3}. B-frag mirrors: N across lane halves, K split across halves.

__device__ __forceinline__ void tile2_gemm_nrm(const float* As, int lda, int K,
                                               const float* __restrict__ W,
                                               int ncol0, int ncol1, int koff,
                                               int ln, v8f& c0, v8f& c1) {
  // B[k][n] = W[k*HIDDEN + n]   (W row-major [K x HIDDEN])
  for (int k = 0; k < K; k += 4) {
    v2f a;
    a.x = As[ln * lda + k + koff];
    a.y = As[ln * lda + k + koff + 1];
    v2f b0, b1;
    b0.x = W[(k + koff) * HIDDEN + ncol0];
    b0.y = W[(k + koff + 1) * HIDDEN + ncol0];
    b1.x = W[(k + koff) * HIDDEN + ncol1];
    b1.y = W[(k + koff + 1) * HIDDEN + ncol1];
    c0 = wmma_f32_k4(a, b0, c0);
    c1 = wmma_f32_k4(a, b1, c1);
  }
}

__device__ __forceinline__ void tile2_gemm_tr(const float* As, int lda, int K,
                                              const float* __restrict__ W,
                                              int ncol0, int ncol1, int koff,
                                              int ln, v8f& c0, v8f& c1) {
  // B[k][n] = W[n*HIDDEN + k]   (multiply by W^T; contiguous float2 along k)
  for (int k = 0; k < K; k += 4) {
    v2f a;
    a.x = As[ln * lda + k + koff];
    a.y = As[ln * lda + k + koff + 1];
    v2f b0, b1;
    b0.x = W[ncol0 * HIDDEN + k + koff];
    b0.y = W[ncol0 * HIDDEN + k + koff + 1];
    b1.x = W[ncol1 * HIDDEN + k + koff];
    b1.y = W[ncol1 * HIDDEN + k + koff + 1];
    c0 = wmma_f32_k4(a, b0, c0);
    c1 = wmma_f32_k4(a, b1, c1);
  }
}

// ---- utility kernels ------------------------------------------------------

__global__ void k_zero(float* __restrict__ p, size_t n) {
  size_t i = (size_t)blockIdx.x * blockDim.x + threadIdx.x;
  if (i < n) p[i] = 0.0f;
}

__global__ void k_deg(const int* __restrict__ src, const int* __restrict__ dst,
                      float* __restrict__ dout, float* __restrict__ din) {
  int e = blockIdx.x * blockDim.x + threadIdx.x;
  if (e < NE) {
    unsafeAtomicAdd(&dout[src[e]], 1.0f);
    unsafeAtomicAdd(&din[dst[e]], 1.0f);
  }
}

__global__ void k_rsqrt(float* __restrict__ a, float* __restrict__ b) {
  int i = blockIdx.x * blockDim.x + threadIdx.x;
  if (i < NN) {
    a[i] = rsqrtf(fmaxf(a[i], 1.0f));
    b[i] = rsqrtf(fmaxf(b[i], 1.0f));
  }
}

// forward scatter: agg[dst] += (x[src] * rsqrt(deg_out[src])) ; 1 wave = 1 edge
__global__ void k_fwd_scatter(const float* __restrict__ q,
                              const float* __restrict__ p,
                              const int* __restrict__ src,
                              const int* __restrict__ dst,
                              const float* __restrict__ rout,
                              float* __restrict__ aggq,
                              float* __restrict__ aggp) {
  size_t tid = (size_t)blockIdx.x * blockDim.x + threadIdx.x;
  size_t e = tid >> 5;
  if (e >= NE) return;
  int lane = (int)(tid & 31);
  int s = src[e], d = dst[e];
  float w = rout[s];
  const float* qs = q + (size_t)s * HIDDEN;
  const float* ps = p + (size_t)s * HIDDEN;
  float* aq = aggq + (size_t)d * HIDDEN;
  float* ap = aggp + (size_t)d * HIDDEN;
#pragma unroll
  for (int t = 0; t < 3; ++t) {
    int j = lane + t * 32;
    unsafeAtomicAdd(&aq[j], qs[j] * w);
    unsafeAtomicAdd(&ap[j], ps[j] * w);
  }
}

// backward scatter (transpose): g[src] += y[dst]
__global__ void k_bwd_scatter(const float* __restrict__ yq,
                              const float* __restrict__ yp,
                              const int* __restrict__ src,
                              const int* __restrict__ dst,
                              float* __restrict__ gq,
                              float* __restrict__ gp) {
  size_t tid = (size_t)blockIdx.x * blockDim.x + threadIdx.x;
  size_t e = tid >> 5;
  if (e >= NE) return;
  int lane = (int)(tid & 31);
  int s = src[e], d = dst[e];
  const float* yqd = yq + (size_t)d * HIDDEN;
  const float* ypd = yp + (size_t)d * HIDDEN;
  float* gqs = gq + (size_t)s * HIDDEN;
  float* gps = gp + (size_t)s * HIDDEN;
#pragma unroll
  for (int t = 0; t < 3; ++t) {
    int j = lane + t * 32;
    unsafeAtomicAdd(&gqs[j], yqd[j]);
    unsafeAtomicAdd(&gps[j], ypd[j]);
  }
}

// q' = q + rsqrt(deg_out)*gp ; p' = p - rsqrt(deg_out)*gq
__global__ void k_finish(const float* __restrict__ q, const float* __restrict__ p,
                         const float* __restrict__ rout,
                         const float* __restrict__ gq, const float* __restrict__ gp,
                         float* __restrict__ outq, float* __restrict__ outp) {
  size_t tid = (size_t)blockIdx.x * blockDim.x + threadIdx.x;
  if (tid >= (size_t)NN * HIDDEN) return;
  int n = (int)(tid / HIDDEN);
  float w = rout[n];
  outq[tid] = q[tid] + w * gp[tid];
  outp[tid] = p[tid] - w * gq[tid];
}

// ---- fused per-node MLP forward+backward, WMMA f32 ------------------------
// 1 wave handles 16 nodes; block = 2 waves; per-wave LDS scratch 24KB.
__global__ __launch_bounds__(64) void k_mlp(
    const float* __restrict__ aggq, const float* __restrict__ aggp,
    const float* __restrict__ rin,
    const float* __restrict__ Wq, const float* __restrict__ bq,
    const float* __restrict__ Wp, const float* __restrict__ bp,
    const float* __restrict__ W1, const float* __restrict__ b1,
    const float* __restrict__ W2, const float* __restrict__ b2,
    float* __restrict__ yq_s, float* __restrict__ yp_s) {
  __shared__ float smem[2 * 6144];  // 48 KB: per-wave zbuf(16x192)+t(16x96)+h(16x96)
  const int wave = threadIdx.x >> 5;
  const int lane = threadIdx.x & 31;
  const int half = lane >> 4;
  const int ln = lane & 15;
  const int koff = half * 2;
  const int node0 = (blockIdx.x * 2 + wave) * 16;
  if (node0 >= NN) return;  // wave-uniform; EXEC stays all-ones for WMMA

  float* zbuf = smem + wave * 6144;  // 16 x 192
  float* tbuf = zbuf + 3072;         // 16 x 96 (t; later overwritten by gu in place)
  float* hbuf = tbuf + 1536;         // 16 x 96

  const float ra = rin[node0 + ln];  // A-row scale (row = ln for both halves)

  // Stage 1+2: qg/pg = (rin .* agg) @ W{q,p} + b  ->  zbuf[:, 0:96 | 96:192]
  for (int fld = 0; fld < 2; ++fld) {
    const float* agg = fld ? aggp : aggq;
    const float* W = fld ? Wp : Wq;
    const float* bias = fld ? bp : bq;
    const float* arow = agg + (size_t)(node0 + ln) * HIDDEN;
    for (int nt = 0; nt < 6; nt += 2) {
      const int ncol0 = nt * 16 + ln;
      const int ncol1 = ncol0 + 16;
      v8f c0 = {}, c1 = {};
      for (int k = 0; k < HIDDEN; k += 4) {
        v2f a;
        a.x = arow[k + koff] * ra;
        a.y = arow[k + koff + 1] * ra;
        v2f b0, b1v;
        b0.x = W[(k + koff) * HIDDEN + ncol0];
        b0.y = W[(k + koff + 1) * HIDDEN + ncol0];
        b1v.x = W[(k + koff) * HIDDEN + ncol1];
        b1v.y = W[(k + koff + 1) * HIDDEN + ncol1];
        c0 = wmma_f32_k4(a, b0, c0);
        c1 = wmma_f32_k4(a, b1v, c1);
      }
      const float bb0 = bias[ncol0];
      const float bb1 = bias[ncol1];
#pragma unroll
      for (int v = 0; v < 8; ++v) {
        zbuf[(v + 8 * half) * 192 + fld * HIDDEN + ncol0] = c0[v] + bb0;
        zbuf[(v + 8 * half) * 192 + fld * HIDDEN + ncol1] = c1[v] + bb1;
      }
    }
  }

  // Stage 3: t = tanh(z @ W1 + b1)
  for (int nt = 0; nt < 6; nt += 2) {
    const int ncol0 = nt * 16 + ln;
    const int ncol1 = ncol0 + 16;
    v8f c0 = {}, c1 = {};
    tile2_gemm_nrm(zbuf, 192, 2 * HIDDEN, W1, ncol0, ncol1, koff, ln, c0, c1);
    const float bb0 = b1[ncol0];
    const float bb1 = b1[ncol1];
#pragma unroll
    for (int v = 0; v < 8; ++v) {
      tbuf[(v + 8 * half) * HIDDEN + ncol0] = tanhf(c0[v] + bb0);
      tbuf[(v + 8 * half) * HIDDEN + ncol1] = tanhf(c1[v] + bb1);
    }
  }

  // Stage 4: h = t @ W2 + b2
  for (int nt = 0; nt < 6; nt += 2) {
    const int ncol0 = nt * 16 + ln;
    const int ncol1 = ncol0 + 16;
    v8f c0 = {}, c1 = {};
    tile2_gemm_nrm(tbuf, HIDDEN, HIDDEN, W2, ncol0, ncol1, koff, ln, c0, c1);
    const float bb0 = b2[ncol0];
    const float bb1 = b2[ncol1];
#pragma unroll
    for (int v = 0; v < 8; ++v) {
      hbuf[(v + 8 * half) * HIDDEN + ncol0] = c0[v] + bb0;
      hbuf[(v + 8 * half) * HIDDEN + ncol1] = c1[v] + bb1;
    }
  }

  // Stage 5: gu = (h @ W2^T) .* (1 - t^2)  -> in place over tbuf
  // (each (m,n) element read/written exactly once by exactly one lane)
  for (int nt = 0; nt < 6; nt += 2) {
    const int ncol0 = nt * 16 + ln;
    const int ncol1 = ncol0 + 16;
    v8f c0 = {}, c1 = {};
    tile2_gemm_tr(hbuf, HIDDEN, HIDDEN, W2, ncol0, ncol1, koff, ln, c0, c1);
#pragma unroll
    for (int v = 0; v < 8; ++v) {
      const int r = (v + 8 * half) * HIDDEN;
      const float t0 = tbuf[r + ncol0];
      const float t1 = tbuf[r + ncol1];
      tbuf[r + ncol0] = c0[v] * (1.0f - t0 * t0);
      tbuf[r + ncol1] = c1[v] * (1.0f - t1 * t1);
    }
  }

  // Stage 6: gz = gu @ W1^T  (16 x 192) -> zbuf
  for (int nt = 0; nt < 12; nt += 2) {
    const int ncol0 = nt * 16 + ln;
    const int ncol1 = ncol0 + 16;
    v8f c0 = {}, c1 = {};
    tile2_gemm_tr(tbuf, HIDDEN, HIDDEN, W1, ncol0, ncol1, koff, ln, c0, c1);
#pragma unroll
    for (int v = 0; v < 8; ++v) {
      zbuf[(v + 8 * half) * 192 + ncol0] = c0[v];
      zbuf[(v + 8 * half) * 192 + ncol1] = c1[v];
    }
  }

  // Stage 7+8: y{q,p} = rin .* (gz_{q,p} @ W{q,p}^T)  -> global
  float r8[8];
#pragma unroll
  for (int v = 0; v < 8; ++v) r8[v] = rin[node0 + v + 8 * half];
  for (int fld = 0; fld < 2; ++fld) {
    const float* W = fld ? Wp : Wq;
    float* yout = fld ? yp_s : yq_s;
    const float* As = zbuf + fld * HIDDEN;  // lda = 192
    for (int nt = 0; nt < 6; nt += 2) {
      const int ncol0 = nt * 16 + ln;
      const int ncol1 = ncol0 + 16;
      v8f c0 = {}, c1 = {};
      tile2_gemm_tr(As, 192, HIDDEN, W, ncol0, ncol1, koff, ln, c0, c1);
#pragma unroll
      for (int v = 0; v < 8; ++v) {
        const size_t row = (size_t)(node0 + v + 8 * half) * HIDDEN;
        yout[row + ncol0] = c0[v] * r8[v];
        yout[row + ncol1] = c1[v] * r8[v];
      }
    }
  }
}

// ---- host launcher --------------------------------------------------------

extern "C" void kernel_launch(void* const* d_in, const int* in_sizes, int n_in,
                              void* d_out, int out_size, void* d_ws, size_t ws_size,
                              hipStream_t stream) {
  const float* q = (const float*)d_in[0];
  const float* p = (const float*)d_in[1];
  const int* src = (const int*)d_in[2];
  const int* dst = (const int*)d_in[3];
  const float* Wq = (const float*)d_in[4];
  const float* bq = (const float*)d_in[5];
  const float* Wp = (const float*)d_in[6];
  const float* bp = (const float*)d_in[7];
  const float* W1 = (const float*)d_in[8];
  const float* b1 = (const float*)d_in[9];
  const float* W2 = (const float*)d_in[10];
  const float* b2 = (const float*)d_in[11];

  const size_t N = NN;
  const size_t NH = (size_t)NN * HIDDEN;

  float* ws = (float*)d_ws;
  float* rout = ws;        // N   : rsqrt(clip(deg_out,1))
  float* rin = rout + N;   // N   : rsqrt(clip(deg_in,1))
  float* aggq = rin + N;   // NH  : fwd accum q ; reused as gq accum in bwd
  float* aggp = aggq + NH; // NH  : fwd accum p ; reused as gp accum in bwd
  float* yq = aggp + NH;   // NH
  float* yp = yq + NH;     // NH

  float* outq = (float*)d_out;
  float* outp = outq + NH;

  // zero degree + fwd-accum buffers (contiguous region)
  {
    size_t n = 2 * N + 2 * NH;
    k_zero<<<(int)((n + 255) / 256), 256, 0, stream>>>(ws, n);
  }
  k_deg<<<(NE + 255) / 256, 256, 0, stream>>>(src, dst, rout, rin);
  k_rsqrt<<<(NN + 255) / 256, 256, 0, stream>>>(rout, rin);
  {
    size_t th = (size_t)NE * 32;
    k_fwd_scatter<<<(int)((th + 255) / 256), 256, 0, stream>>>(q, p, src, dst,
                                                               rout, aggq, aggp);
  }
  {
    int nwaves = (NN + 15) / 16;  // 3125
    int blocks = (nwaves + 1) / 2;
    k_mlp<<<blocks, 64, 0, stream>>>(aggq, aggp, rin, Wq, bq, Wp, bp, W1, b1,
                                     W2, b2, yq, yp);
  }
  {  // re-zero aggq/aggp as backward accumulators
    size_t n = 2 * NH;
    k_zero<<<(int)((n + 255) / 256), 256, 0, stream>>>(aggq, n);
  }
  {
    size_t th = (size_t)NE * 32;
    k_bwd_scatter<<<(int)((th + 255) / 256), 256, 0, stream>>>(yq, yp, src, dst,
                                                               aggq, aggp);
  }
  k_finish<<<(int)((NH + 255) / 256), 256, 0, stream>>>(q, p, rout, aggq, aggp,
                                                        outq, outp);
}